// SignalingNetworkPredictor_26525718020342
// MI455X (gfx1250) — compile-verified
//
#include <hip/hip_runtime.h>
#include <hip/hip_bf16.h>
#include <cmath>

typedef __attribute__((ext_vector_type(2))) float v2f;
typedef __attribute__((ext_vector_type(8))) float v8f;

#define N_TOK 1024
#define DHID  128

// ---------------------------------------------------------------------------
// Kernel 1: fused linear projections  Q = z@Wq+bq, K = z@Wk+bk,
//           Ap = z@W1[:D], Bp = z@W1[D:]   (all [1024,128], f32 WMMA)
// One wave (32 lanes) computes one 16x16 output tile via 32x V_WMMA_F32_16X16X4_F32.
// ---------------------------------------------------------------------------
__global__ __launch_bounds__(128) void snp_proj_gemm(
    const float* __restrict__ z,
    const float* __restrict__ Wq, const float* __restrict__ bq,
    const float* __restrict__ Wk, const float* __restrict__ bk,
    const float* __restrict__ W1,
    float* __restrict__ Qo, float* __restrict__ Ko,
    float* __restrict__ Ao, float* __restrict__ Bo)
{
    const int lane   = threadIdx.x & 31;
    const int waveId = blockIdx.x * 4 + (threadIdx.x >> 5);
    const int matIdx = waveId >> 9;        // 512 tiles per matrix
    const int t      = waveId & 511;
    const int tileM  = t >> 3;             // 0..63
    const int tileN  = t & 7;              // 0..7

    const float* W; const float* bias; float* out;
    if (matIdx == 0)      { W = Wq;              bias = bq;      out = Qo; }
    else if (matIdx == 1) { W = Wk;              bias = bk;      out = Ko; }
    else if (matIdx == 2) { W = W1;              bias = nullptr; out = Ao; }
    else                  { W = W1 + DHID*DHID;  bias = nullptr; out = Bo; }

    const int m    = lane & 15;            // M for A-frag / N for B,C,D frags
    const int half = lane >> 4;
    const int rowA = tileM * 16 + m;
    const int colB = tileN * 16 + m;

    v8f c = {};
    for (int k = 0; k < DHID; k += 4) {
        const int kk = k + 2 * half;
        v2f a, b;
        a.x = z[rowA * DHID + kk];
        a.y = z[rowA * DHID + kk + 1];
        b.x = W[kk * DHID + colB];
        b.y = W[(kk + 1) * DHID + colB];
        c = __builtin_amdgcn_wmma_f32_16x16x4_f32(false, a, false, b,
                                                  (short)0, c, false, false);
    }
    const float bv = bias ? bias[colB] : 0.0f;
#pragma unroll
    for (int r = 0; r < 8; ++r) {
        const int row = tileM * 16 + r + 8 * half;   // C/D layout: M = r (+8 for upper half)
        out[row * DHID + colB] = c[r] + bv;
    }
}

// ---------------------------------------------------------------------------
// Kernel 2: attention strip (16 rows) -> WMMA Q.K^T * scale * exp(-dist) into
// LDS, wave-local softmax + iterative top-32 extraction, sparse row writeout.
// Block = 256 threads (8 waves), one block per 16-row strip.
// ---------------------------------------------------------------------------
__global__ __launch_bounds__(256) void snp_attn_topk(
    const float* __restrict__ Q, const float* __restrict__ K,
    const float* __restrict__ dist, float* __restrict__ adj)
{
    __shared__ float at[16][N_TOK];   // 64 KB of the 320 KB WGP LDS

    const int lane = threadIdx.x & 31;
    const int w    = threadIdx.x >> 5;
    const int i0   = blockIdx.x * 16;
    const int m    = lane & 15;
    const int half = lane >> 4;
    const float scale = 0.08838834764831845f;   // 1/sqrt(128), TEMPERATURE = 1

    // --- raw attention tiles: wave w covers j-tiles {w, w+8, ..., w+56} ---
    for (int s = 0; s < 8; ++s) {
        const int j0 = (w + 8 * s) * 16;
        v8f c = {};
        for (int k = 0; k < DHID; k += 4) {
            const int kk = k + 2 * half;
            v2f a, b;
            a.x = Q[(i0 + m) * DHID + kk];      // A = Q tile (row-major)
            a.y = Q[(i0 + m) * DHID + kk + 1];
            b.x = K[(j0 + m) * DHID + kk];      // B = K^T tile: col n is K row j0+n
            b.y = K[(j0 + m) * DHID + kk + 1];
            c = __builtin_amdgcn_wmma_f32_16x16x4_f32(false, a, false, b,
                                                      (short)0, c, false, false);
        }
#pragma unroll
        for (int r = 0; r < 8; ++r) {
            const int il = r + 8 * half;
            const int j  = j0 + m;
            at[il][j] = c[r] * scale * expf(-dist[(i0 + il) * N_TOK + j]);
        }
    }
    __syncthreads();

    // --- each wave owns rows 2w and 2w+1: softmax + top-32 + writeout ---
    for (int rr = 0; rr < 2; ++rr) {
        const int row = 2 * w + rr;
        float* rv = at[row];

        // row max (lane owns cols lane + 32t)
        float mx = -INFINITY;
        for (int tt = 0; tt < 32; ++tt) mx = fmaxf(mx, rv[lane + 32 * tt]);
        for (int off = 16; off; off >>= 1) mx = fmaxf(mx, __shfl_xor(mx, off, 32));

        // exp + sum, normalize in place
        float ev[32];
        float sum = 0.0f;
        for (int tt = 0; tt < 32; ++tt) {
            const float e = expf(rv[lane + 32 * tt] - mx);
            ev[tt] = e;
            sum += e;
        }
        for (int off = 16; off; off >>= 1) sum += __shfl_xor(sum, off, 32);
        const float inv = 1.0f / sum;
        for (int tt = 0; tt < 32; ++tt) rv[lane + 32 * tt] = ev[tt] * inv;

        // iterative top-32 with min-index tie-break (matches lax.top_k set)
        unsigned sel = 0u;   // bit tt -> col (lane + 32*tt) selected
        for (int it = 0; it < 32; ++it) {
            float bv = -1.0f;
            int   bi = 1 << 30;
            for (int tt = 0; tt < 32; ++tt) {
                if ((sel >> tt) & 1u) continue;
                const float vv = rv[lane + 32 * tt];
                if (vv > bv) { bv = vv; bi = lane + 32 * tt; }   // strict > keeps min index
            }
            for (int off = 16; off; off >>= 1) {
                const float ov = __shfl_xor(bv, off, 32);
                const int   oi = __shfl_xor(bi, off, 32);
                if (ov > bv || (ov == bv && oi < bi)) { bv = ov; bi = oi; }
            }
            if ((bi & 31) == lane) sel |= 1u << (bi >> 5);
        }

        // sparse row writeout: keep selected softmax values, zero elsewhere
        for (int tt = 0; tt < 32; ++tt) {
            const int col = lane + 32 * tt;
            adj[(i0 + row) * N_TOK + col] = ((sel >> tt) & 1u) ? rv[col] : 0.0f;
        }
    }
}

// ---------------------------------------------------------------------------
// Kernel 3: pairwise classifier. 16x16 pair tile per 256-thread block.
// h = gelu(a_i + b_j + b1) (exact erf), logits = h @ W2 + b2, softmax over 3.
// ---------------------------------------------------------------------------
__global__ __launch_bounds__(256) void snp_pairwise(
    const float* __restrict__ Ap, const float* __restrict__ Bp,
    const float* __restrict__ b1, const float* __restrict__ W2,
    const float* __restrict__ b2, float* __restrict__ outp)
{
    __shared__ float As[16][DHID + 2];   // +2 pad: kills 2-way LDS bank conflict
    __shared__ float Bs[16][DHID + 2];
    __shared__ float b1s[DHID];
    __shared__ float w2s[DHID * 3];
    __shared__ float b2s[3];

    const int ib  = blockIdx.x >> 6;     // 64 x 64 tile grid
    const int jb  = blockIdx.x & 63;
    const int tid = threadIdx.x;

    for (int idx = tid; idx < 16 * DHID; idx += 256) {
        const int r = idx >> 7, d = idx & 127;
        As[r][d] = Ap[(ib * 16 + r) * DHID + d];
        Bs[r][d] = Bp[(jb * 16 + r) * DHID + d];
    }
    for (int idx = tid; idx < DHID; idx += 256)     b1s[idx] = b1[idx];
    for (int idx = tid; idx < DHID * 3; idx += 256) w2s[idx] = W2[idx];
    if (tid < 3) b2s[tid] = b2[tid];
    __syncthreads();

    const int il = tid >> 4, jl = tid & 15;
    float acc0 = 0.0f, acc1 = 0.0f, acc2 = 0.0f;
#pragma unroll 4
    for (int d = 0; d < DHID; ++d) {
        const float x = As[il][d] + Bs[jl][d] + b1s[d];
        const float g = 0.5f * x * (1.0f + erff(x * 0.70710678118654752f)); // exact gelu
        acc0 += g * w2s[d * 3 + 0];
        acc1 += g * w2s[d * 3 + 1];
        acc2 += g * w2s[d * 3 + 2];
    }
    acc0 += b2s[0]; acc1 += b2s[1]; acc2 += b2s[2];

    const float mx = fmaxf(acc0, fmaxf(acc1, acc2));
    const float e0 = expf(acc0 - mx);
    const float e1 = expf(acc1 - mx);
    const float e2 = expf(acc2 - mx);
    const float inv = 1.0f / (e0 + e1 + e2);

    const long long p = (long long)(ib * 16 + il) * N_TOK + (jb * 16 + jl);
    float* o = outp + p * 3;
    o[0] = e0 * inv; o[1] = e1 * inv; o[2] = e2 * inv;
}

// ---------------------------------------------------------------------------
extern "C" void kernel_launch(void* const* d_in, const int* in_sizes, int n_in,
                              void* d_out, int out_size, void* d_ws, size_t ws_size,
                              hipStream_t stream) {
    (void)in_sizes; (void)n_in; (void)out_size; (void)ws_size;

    const float* z    = (const float*)d_in[0];
    const float* dist = (const float*)d_in[1];
    const float* Wq   = (const float*)d_in[2];
    const float* bq   = (const float*)d_in[3];
    const float* Wk   = (const float*)d_in[4];
    const float* bk   = (const float*)d_in[5];
    const float* W1   = (const float*)d_in[6];
    const float* b1   = (const float*)d_in[7];
    const float* W2   = (const float*)d_in[8];
    const float* b2   = (const float*)d_in[9];

    float* adj = (float*)d_out;                     // [1024,1024]
    float* sig = (float*)d_out + N_TOK * N_TOK;     // [1024,1024,3]

    float* ws = (float*)d_ws;                       // 4 x [1024,128] f32 = 2 MB
    float* Q  = ws;
    float* K  = ws + 1 * N_TOK * DHID;
    float* Ap = ws + 2 * N_TOK * DHID;
    float* Bp = ws + 3 * N_TOK * DHID;

    // 1) projections: 2048 tiles, 4 waves/block -> 512 blocks
    snp_proj_gemm<<<512, 128, 0, stream>>>(z, Wq, bq, Wk, bk, W1, Q, K, Ap, Bp);

    // 2) attention + softmax + top-32: one block per 16-row strip
    snp_attn_topk<<<N_TOK / 16, 256, 0, stream>>>(Q, K, dist, adj);

    // 3) pairwise GELU classifier: 64x64 pair tiles
    snp_pairwise<<<(N_TOK / 16) * (N_TOK / 16), 256, 0, stream>>>(Ap, Bp, b1, W2, b2, sig);
}